// CustomRNN_4715874091406
// MI455X (gfx1250) — compile-verified
//
#include <hip/hip_runtime.h>
#include <math.h>

// CustomRNN forward for MI455X (gfx1250), fp32 end-to-end using
// V_WMMA_F32_16X16X4_F32 (exact fp32 math, matching the reference).
//
// Phase 1: xproj = X @ W_ih^T + b_ih  (32768x1024x1024 GEMM, LDS-tiled WMMA,
//          double-buffered GLOBAL_LOAD_ASYNC_TO_LDS staging on ASYNCcnt).
// Phase 2: persistent 32-WG kernel, W_hh slice resident in LDS (128KB/WG),
//          global-atomic grid barrier between the 512 sequential steps,
//          h_t written in place over xproj[t] in d_out.
//
// LDS layouts are fragment-native so every WMMA operand is a single aligned
// ds_load_b64:
//   B tiles: k-pairs interleaved as float2 per column; the 2nd pair of each
//   k-quad is stored at column (n^16) so the two half-waves (which need k and
//   k+2) hit disjoint 32-bank halves -> conflict-free.
//   A tile:  row stride 36 floats -> 16B-aligned b128 staging and 8B-aligned
//   b64 fragment reads, conflict-free (4*(9r mod 16) injective).

#define T_STEPS 512
#define BATCH   64
#define DIM     1024

typedef float v2f __attribute__((ext_vector_type(2)));
typedef float v4f __attribute__((ext_vector_type(4)));
typedef float v8f __attribute__((ext_vector_type(8)));

static __device__ __forceinline__ v8f v8f_zero() {
  v8f z = {0.f, 0.f, 0.f, 0.f, 0.f, 0.f, 0.f, 0.f};
  return z;
}

// Async Global->LDS copies (CDNA5 GLOBAL_LOAD_ASYNC_TO_LDS_*, ASYNCcnt).
// LDS destination = low 32 bits of the generic pointer (ISA: LDS flat
// addresses truncate to addr[31:0]).
static __device__ __forceinline__ void async_copy_b128(unsigned lds_off,
                                                       const float* gaddr) {
  asm volatile("global_load_async_to_lds_b128 %0, %1, off"
               :: "v"(lds_off), "v"(gaddr) : "memory");
}
static __device__ __forceinline__ void async_copy_b64(unsigned lds_off,
                                                      const float* gaddr) {
  asm volatile("global_load_async_to_lds_b64 %0, %1, off"
               :: "v"(lds_off), "v"(gaddr) : "memory");
}

// ---------------------------------------------------------------------------
// Phase 1: xproj[M=T*B, N=D] = X[M,K=D] @ W_ih^T + b_ih
// Workgroup tile: 128(M) x 64(N), K in chunks of 32, double-buffered async
// staging. 8 waves: wave w owns M-stripe [w*16, w*16+16), all 64 N columns.
// ---------------------------------------------------------------------------
#define KC 32
#define XS_STRIDE 36   // 128 rows x 36 floats (pad 32 -> 36)

__global__ __launch_bounds__(256) void rnn_xproj_kernel(
    const float* __restrict__ x,      // [T*B, D]
    const float* __restrict__ W,      // [D, D] row-major (h, d)
    const float* __restrict__ bias,   // [D]
    float* __restrict__ out)          // [T*B, D]
{
  __shared__ float Xs[2][128 * XS_STRIDE];     // 2 x 18.4 KB
  __shared__ float Wq[2][KC * 64];             // 2 x 8 KB, pair-swizzled

  const int tid    = threadIdx.x;
  const int lane   = tid & 31;
  const int w      = tid >> 5;
  const int lane15 = lane & 15;
  const int h      = lane >> 4;                // half-wave select
  const int Mbase  = blockIdx.y * 128;
  const int Nbase  = blockIdx.x * 64;

  v8f acc[4];
  acc[0] = v8f_zero(); acc[1] = v8f_zero(); acc[2] = v8f_zero(); acc[3] = v8f_zero();

  const int arow = (w << 4) + lane15;          // A-frag row (M within tile)
  const int khi  = h << 1;                     // lanes 16-31 hold K+2, K+3

  // Per-thread staging geometry (constant across chunks).
  const int xr  = tid >> 3;                    // X rows: 4 per thread (j adds 32)
  const int xc4 = (tid & 7) << 2;
  const int wn  = tid >> 3;                    // W rows handled (f>>3), j adds 32
  const int wc4 = (tid & 7) << 2;
  const int wqc = wc4 >> 2;

  // Issue one chunk's staging: 8 async ops per thread (in-order completion).
  auto issue_stage = [&](int buf, int k0) {
#pragma unroll
    for (int j = 0; j < 4; ++j) {              // X: 1024 quads, b128 each
      const int r = xr + j * 32;
      async_copy_b128(
          (unsigned)(size_t)&Xs[buf][r * XS_STRIDE + xc4],
          x + (size_t)(Mbase + r) * DIM + (k0 + xc4));
    }
#pragma unroll
    for (int j = 0; j < 2; ++j) {              // W: 512 quads, 2 x b64 each
      const int n = wn + j * 32;
      const float* g = W + (size_t)(Nbase + n) * DIM + (k0 + wc4);
      async_copy_b64(
          (unsigned)(size_t)&Wq[buf][(((wqc << 1) + 0) * 64 + n) << 1], g);
      async_copy_b64(
          (unsigned)(size_t)&Wq[buf][(((wqc << 1) + 1) * 64 + (n ^ 16)) << 1],
          g + 2);
    }
  };

  issue_stage(0, 0);
  for (int k0 = 0, it = 0; k0 < DIM; k0 += KC, ++it) {
    const int cur = it & 1;
    const bool more = (k0 + KC) < DIM;
    if (more) {
      // Prefetch next chunk into the other buffer (safe: last reader of that
      // buffer passed the end-of-iteration barrier one iteration ago).
      issue_stage(cur ^ 1, k0 + KC);
      // In-order completion: <=8 outstanding means current chunk has landed.
      asm volatile("s_wait_asynccnt 0x8" ::: "memory");
    } else {
      asm volatile("s_wait_asynccnt 0x0" ::: "memory");
    }
    __syncthreads();

    const v2f* Wq2 = (const v2f*)&Wq[cur][0];
    const float* XsC = &Xs[cur][0];
#pragma unroll
    for (int kk = 0; kk < KC; kk += 4) {
      const int qc = kk >> 2;
      v2f a = *(const v2f*)(XsC + arow * XS_STRIDE + kk + khi);  // aligned b64
#pragma unroll
      for (int jj = 0; jj < 4; ++jj) {
        const int col = (jj << 4) + lane15;
        const int cS  = h ? (col ^ 16) : col;
        v2f b = Wq2[((qc << 1) + h) * 64 + cS];                  // aligned b64
        acc[jj] = __builtin_amdgcn_wmma_f32_16x16x4_f32(
            false, a, false, b, (short)0, acc[jj], false, false);
      }
    }
    __syncthreads();   // all waves done reading buffer `cur` before it is refilled
  }

  // Epilogue: C layout — VGPR i: lanes 0-15 -> M=i, lanes 16-31 -> M=i+8.
  const int rbase = Mbase + (w << 4) + (h << 3);
#pragma unroll
  for (int jj = 0; jj < 4; ++jj) {
    const int col = Nbase + (jj << 4) + lane15;
    const float bv = bias[col];
#pragma unroll
    for (int i = 0; i < 8; ++i)
      out[(size_t)(rbase + i) * DIM + col] = acc[jj][i] + bv;
  }
}

// ---------------------------------------------------------------------------
// Phase 2: persistent recurrence kernel.
//   NWG=32 workgroups, each owns NS=32 hidden columns.
//   W_hh slice lives in dynamic LDS in the pair-swizzled B layout:
//   DIM x NS floats = 128KB per WG (2 WGs per 320KB WGP).
//   Per step: h_t[b, cols] = tanh(xproj[t] + h_{t-1} @ Whh^T + b_hh), in place
//   over d_out[t]. Grid barrier (monotonic global atomic) between steps.
// ---------------------------------------------------------------------------
#define NWG 32
#define NS  32

__global__ __launch_bounds__(256) void rnn_recurrence_kernel(
    const float* __restrict__ Whh,   // [D, D]
    const float* __restrict__ bhh,   // [D]
    float* __restrict__ out,         // [T, B, D]; xproj in, h out (in place)
    unsigned int* __restrict__ barrier_cnt)
{
  extern __shared__ float Wl[];      // DIM*NS floats, pair-swizzled
  const int tid    = threadIdx.x;
  const int lane   = tid & 31;
  const int w      = tid >> 5;
  const int lane15 = lane & 15;
  const int h      = lane >> 4;
  const int Nbase  = blockIdx.x * NS;

  // One-time load of this WG's W_hh slice (float4 global reads, coalesced),
  // scattered into the pair-swizzled layout.
#pragma unroll 4
  for (int j = 0; j < (NS * DIM / 4) / 256; ++j) {   // 32 iterations
    const int f = tid + j * 256;                     // quad index, 8192 total
    const int n = f >> 8;                            // 256 quads per row
    const int q = f & 255;
    v4f v = *(const v4f*)(Whh + (size_t)(Nbase + n) * DIM + (q << 2));
    v2f p0 = {v.x, v.y};
    v2f p1 = {v.z, v.w};
    *(v2f*)(&Wl[(((q << 1) + 0) * NS + n) << 1])        = p0;
    *(v2f*)(&Wl[(((q << 1) + 1) * NS + (n ^ 16)) << 1]) = p1;
  }
  __syncthreads();
  const v2f* Wl2 = (const v2f*)Wl;

  // 8 waves = 4 M-stripes (16 batch rows) x 2 N-halves (16 cols): one 16x16
  // C tile per wave.
  const int wm    = w & 3;
  const int wvn   = w >> 2;
  const int arow  = (wm << 4) + lane15;             // batch row for A frag
  const int khi   = h << 1;
  const int lcol  = (wvn << 4) + lane15;            // local column 0..31
  const int cS    = h ? (lcol ^ 16) : lcol;         // swizzled B column
  const int gcol  = Nbase + lcol;                   // global hidden index
  const float bv  = bhh[gcol];
  const int rbase = (wm << 4) + (h << 3);           // C-tile row base

  for (int t = 0; t < T_STEPS; ++t) {
    float* xp = out + (size_t)t * BATCH * DIM;
    v8f acc0 = v8f_zero();
    v8f acc1 = v8f_zero();

    if (t > 0) {
      const float* hrow = out + (size_t)(t - 1) * BATCH * DIM + (size_t)arow * DIM;
      // Two interleaved accumulators break the WMMA D->C RAW chain.
#pragma unroll 4
      for (int k = 0; k < DIM; k += 8) {
        const int q0 = k >> 2;
        v2f a0 = *(const v2f*)(hrow + k + khi);          // aligned b64
        v2f b0 = Wl2[((q0 << 1) + h) * NS + cS];         // aligned b64
        acc0 = __builtin_amdgcn_wmma_f32_16x16x4_f32(
            false, a0, false, b0, (short)0, acc0, false, false);
        v2f a1 = *(const v2f*)(hrow + k + 4 + khi);
        v2f b1 = Wl2[(((q0 + 1) << 1) + h) * NS + cS];
        acc1 = __builtin_amdgcn_wmma_f32_16x16x4_f32(
            false, a1, false, b1, (short)0, acc1, false, false);
      }
    }

    v8f acc = acc0 + acc1;
#pragma unroll
    for (int i = 0; i < 8; ++i) {
      const size_t idx = (size_t)(rbase + i) * DIM + gcol;
      xp[idx] = tanhf(acc[i] + xp[idx] + bv);   // read xproj, write h (same elem)
    }

    // ---- device-wide step barrier (monotonic counter, no reset needed) ----
    __threadfence();          // make this thread's h stores device-visible
    __syncthreads();          // all waves in WG done with step t
    if (tid == 0) {
      __hip_atomic_fetch_add(barrier_cnt, 1u, __ATOMIC_RELEASE,
                             __HIP_MEMORY_SCOPE_AGENT);
      const unsigned target = (unsigned)(t + 1) * NWG;
      while (__hip_atomic_load(barrier_cnt, __ATOMIC_ACQUIRE,
                               __HIP_MEMORY_SCOPE_AGENT) < target)
        __builtin_amdgcn_s_sleep(2);
    }
    __syncthreads();
    __threadfence();          // acquire: invalidate stale WGP-cache lines of h
  }
}

// ---------------------------------------------------------------------------
extern "C" void kernel_launch(void* const* d_in, const int* in_sizes, int n_in,
                              void* d_out, int out_size, void* d_ws, size_t ws_size,
                              hipStream_t stream) {
  (void)in_sizes; (void)n_in; (void)out_size; (void)ws_size;
  const float* x    = (const float*)d_in[0];   // [T, B, D]
  const float* W_ih = (const float*)d_in[1];   // [D, D]
  const float* W_hh = (const float*)d_in[2];   // [D, D]
  const float* b_ih = (const float*)d_in[3];   // [D]
  const float* b_hh = (const float*)d_in[4];   // [D]
  float* out = (float*)d_out;                  // [T*B*D] output + [B*D] final h

  // Zero the grid-barrier counter (capture-safe).
  hipMemsetAsync(d_ws, 0, sizeof(unsigned int), stream);

  // Phase 1: xproj -> d_out (in-place consumed by phase 2).
  dim3 g1(DIM / 64, (T_STEPS * BATCH) / 128);
  rnn_xproj_kernel<<<g1, 256, 0, stream>>>(x, W_ih, b_ih, out);

  // Phase 2: persistent recurrence, 128KB dynamic LDS per WG.
  const size_t lds_bytes = (size_t)DIM * NS * sizeof(float);
  rnn_recurrence_kernel<<<NWG, 256, lds_bytes, stream>>>(
      W_hh, b_hh, out, (unsigned int*)d_ws);

  // Final hidden state = h_{T-1}, appended after the [T,B,D] output.
  hipMemcpyAsync(out + (size_t)T_STEPS * BATCH * DIM,
                 out + (size_t)(T_STEPS - 1) * BATCH * DIM,
                 (size_t)BATCH * DIM * sizeof(float),
                 hipMemcpyDeviceToDevice, stream);
}